// SingleAttentionNodeLayer_41850161332541
// MI455X (gfx1250) — compile-verified
//
#include <hip/hip_runtime.h>
#include <hip/hip_bf16.h>
#include <math.h>

// ---------------------------------------------------------------------------
// GAT single-attention layer, algebraically factored:
//   w1 = W @ a1                         (256)
//   c[b] = (centernode[b] @ W) . a2     (64)
//   e[b,n] = leaky_relu(X[b,n,:].w1 + c[b]); masked by A>0 else -1e12
//   out[b,:] = softmax_n(e[b,:]) @ X[b]   (streamed, chunked online softmax)
// X read ~once (256 MB) -> HBM-bound, ~11us floor on MI455X (23.3 TB/s).
// Dot products use V_WMMA_F32_16X16X4_F32 (fp32-exact).
// ---------------------------------------------------------------------------

typedef __attribute__((ext_vector_type(2))) float v2f;
typedef __attribute__((ext_vector_type(4))) float v4f;
typedef __attribute__((ext_vector_type(8))) float v8f;

#define BATCH    64
#define NROWS    4096
#define NF       256
#define CHUNKS   32          // chunks per batch
#define CHROWS   128         // rows per chunk (8 waves * 16 rows)
#define NEG_INF  -1000000000000.0f
#define REC_SZ   258         // m, s, v[256]
#define PART_OFF 320         // ws float offset of partial records

// ---- Kernel 1: w1 = W@a1 ; c[b] = centernode[b] . (W@a2) --------------------
__global__ void __launch_bounds__(256)
gat_prep(const float* __restrict__ cn, const float* __restrict__ W,
         const float* __restrict__ a, float* __restrict__ ws) {
  __shared__ float s_w2[NF];
  int tid = threadIdx.x;                    // 256 threads
  float w1 = 0.f, w2 = 0.f;
  const float* wr = W + (size_t)tid * NF;   // row tid of W
  #pragma unroll 4
  for (int j = 0; j < NF; ++j) {
    float wv = wr[j];
    w1 += wv * a[j];
    w2 += wv * a[NF + j];
  }
  ws[tid]   = w1;                           // ws[0..255] = w1
  s_w2[tid] = w2;
  __syncthreads();
  if (tid < BATCH) {
    const float* cb = cn + (size_t)tid * NF;
    float c = 0.f;
    #pragma unroll 4
    for (int i = 0; i < NF; ++i) c += cb[i] * s_w2[i];
    ws[NF + tid] = c;                       // ws[256..319] = c[b]
  }
}

// ---- Kernel 2: per-chunk scores (WMMA), softmax partials, weighted sum -----
__global__ void __launch_bounds__(256)
gat_main(const float* __restrict__ X, const float* __restrict__ A,
         const float* __restrict__ ws, float* __restrict__ partials) {
  __shared__ float s_w1[NF];
  __shared__ float s_e[CHROWS];
  __shared__ float s_red[64];
  __shared__ float s_v[NF];

  const int tid   = threadIdx.x;
  const int lane  = tid & 31;
  const int wave  = tid >> 5;               // 0..7
  const int b     = blockIdx.x / CHUNKS;
  const int chunk = blockIdx.x % CHUNKS;
  const int row0  = chunk * CHROWS;         // first row of this chunk in batch b

  const float* Xb = X + (size_t)b * NROWS * NF;

  s_w1[tid] = ws[tid];                      // stage w1 in LDS
  s_v[tid]  = 0.f;
  __syncthreads();

  // ---- Phase 1: e-dots for 16 rows per wave via V_WMMA_F32_16X16X4_F32 ----
  // A (16x4 f32): VGPR0 = {K0 | K2}, VGPR1 = {K1 | K3}; lanes 0-15 -> M=lane.
  // B (4x16 f32): same half split; all 16 columns hold the same w1 values.
  const int half = lane >> 4;               // 0: K0/K1 half, 1: K2/K3 half
  const int mrow = lane & 15;
  const int wrow = row0 + wave * 16;        // wave's 16 rows
  const float* xrow = Xb + (size_t)(wrow + mrow) * NF + half * 2;

  v8f acc = {};
  #pragma unroll 8
  for (int k0 = 0; k0 < NF; k0 += 4) {
    v2f av = *(const v2f*)(xrow + k0);
    v2f bv = *(const v2f*)(&s_w1[k0 + half * 2]);
    acc = __builtin_amdgcn_wmma_f32_16x16x4_f32(
        /*neg_a=*/false, av, /*neg_b=*/false, bv,
        /*c_mod=*/(short)0, acc, /*reuse_a=*/false, /*reuse_b=*/false);
  }
  // D column 0: lane 0 holds rows M=0..7 in acc[0..7], lane 16 holds M=8..15.
  if (mrow == 0) {
    int rb = wave * 16 + half * 8;
    #pragma unroll
    for (int i = 0; i < 8; ++i) s_e[rb + i] = acc[i];
  }
  __syncthreads();

  // ---- finalize e: + c_b, leaky_relu(0.01), edge mask ----
  if (tid < CHROWS) {
    float e = s_e[tid] + ws[NF + b];
    e = (e > 0.f) ? e : 0.01f * e;
    float Av = A[(size_t)b * NROWS + row0 + tid];
    s_e[tid] = (Av > 0.f) ? e : NEG_INF;
  }
  __syncthreads();

  // ---- chunk max ----
  if (tid < 64) s_red[tid] = fmaxf(s_e[tid], s_e[tid + 64]);
  __syncthreads();
  for (int off = 32; off >= 1; off >>= 1) {
    if (tid < off) s_red[tid] = fmaxf(s_red[tid], s_red[tid + off]);
    __syncthreads();
  }
  const float m = s_red[0];
  __syncthreads();

  // ---- p = exp(e - m) ----
  if (tid < CHROWS) s_e[tid] = __expf(s_e[tid] - m);
  __syncthreads();

  // ---- chunk sum ----
  if (tid < 64) s_red[tid] = s_e[tid] + s_e[tid + 64];
  __syncthreads();
  for (int off = 32; off >= 1; off >>= 1) {
    if (tid < off) s_red[tid] += s_red[tid + off];
    __syncthreads();
  }

  // ---- Phase 3: v = sum_r p[r] * X[row_r,:]  (rows are L0/L2 hot) ----
  // lane j owns cols {4j..4j+3} and {128+4j..128+4j+3}.
  float vac[8] = {0.f, 0.f, 0.f, 0.f, 0.f, 0.f, 0.f, 0.f};
  const float* xr2 = Xb + (size_t)wrow * NF;
  #pragma unroll 4
  for (int r = 0; r < 16; ++r) {
    float p = s_e[wave * 16 + r];
    const float* rp = xr2 + r * NF;
    v4f x0 = *(const v4f*)(rp + lane * 4);
    v4f x1 = *(const v4f*)(rp + 128 + lane * 4);
    #pragma unroll
    for (int i = 0; i < 4; ++i) { vac[i] += p * x0[i]; vac[4 + i] += p * x1[i]; }
  }
  #pragma unroll
  for (int i = 0; i < 4; ++i) {
    atomicAdd(&s_v[lane * 4 + i],       vac[i]);      // ds_add_f32
    atomicAdd(&s_v[128 + lane * 4 + i], vac[4 + i]);
  }
  __syncthreads();

  // ---- emit partial record {m, s, v[256]} ----
  float* rec = partials + (size_t)(b * CHUNKS + chunk) * REC_SZ;
  if (tid == 0) { rec[0] = m; rec[1] = s_red[0]; }
  rec[2 + tid] = s_v[tid];
}

// ---- Kernel 3: combine 32 chunk partials per batch -------------------------
__global__ void __launch_bounds__(256)
gat_finalize(const float* __restrict__ partials, float* __restrict__ out) {
  const int b = blockIdx.x;
  const int j = threadIdx.x;
  const float* base = partials + (size_t)b * CHUNKS * REC_SZ;

  float m = -INFINITY;
  for (int i = 0; i < CHUNKS; ++i) m = fmaxf(m, base[(size_t)i * REC_SZ]);

  float s = 0.f, v = 0.f;
  for (int i = 0; i < CHUNKS; ++i) {
    const float* rec = base + (size_t)i * REC_SZ;
    float sc = __expf(rec[0] - m);
    s += rec[1] * sc;
    v += rec[2 + j] * sc;
  }
  out[(size_t)b * NF + j] = v / s;
}

// ---------------------------------------------------------------------------
extern "C" void kernel_launch(void* const* d_in, const int* in_sizes, int n_in,
                              void* d_out, int out_size, void* d_ws, size_t ws_size,
                              hipStream_t stream) {
  const float* X  = (const float*)d_in[0];   // (64, 4096, 256)
  const float* A  = (const float*)d_in[1];   // (64, 4096, 1)
  const float* cn = (const float*)d_in[2];   // (64, 1, 256)
  const float* W  = (const float*)d_in[3];   // (256, 256)
  const float* a  = (const float*)d_in[4];   // (512, 1)
  float* out = (float*)d_out;                // (64, 1, 256)
  float* ws  = (float*)d_ws;

  gat_prep<<<1, 256, 0, stream>>>(cn, W, a, ws);
  gat_main<<<BATCH * CHUNKS, 256, 0, stream>>>(X, A, ws, ws + PART_OFF);
  gat_finalize<<<BATCH, 256, 0, stream>>>(ws + PART_OFF, out);
}